// FCOSPostProcessor_25683904430604
// MI455X (gfx1250) — compile-verified
//
#include <hip/hip_runtime.h>
#include <hip/hip_bf16.h>
#include <math.h>

// ---------------- problem constants (from setup_inputs) ----------------
#define NIMG   16
#define NCLS   80
#define IMH    200
#define IMW    256
#define HWPIX  (IMH * IMW)          // 51200
#define CHW    (NCLS * HWPIX)       // 4,096,000 per image
#define KPAD   1024                 // padded top-k slots (reference uses 1000)
#define KREAL  1000
#define BINS   1024
#define CAP    4096                 // candidate gather capacity per image
#define OUTN   100
#define NEGV   (-1000000000.0f)
#define CLSOFF 4096.0f
#define NMSTH  0.6f

typedef float v2f __attribute__((ext_vector_type(2)));
typedef float v8f __attribute__((ext_vector_type(8)));

// ---------------- pass 1: per-image score histogram (float4 streaming) ----------------
__global__ void k_hist(const float* __restrict__ cls,
                       const float* __restrict__ ctr,
                       int* __restrict__ ghist) {
  __shared__ int lh[BINS];
  const int img = blockIdx.y;
  const int tid = threadIdx.x;                      // 256 threads
  for (int b = tid; b < BINS; b += 256) lh[b] = 0;
  __syncthreads();
  const float4* cb4 = (const float4*)(cls + (size_t)img * CHW);
  const float*  tb  = ctr + (size_t)img * HWPIX;
  const int q0 = blockIdx.x * 4096;                 // float4 index base (16384 elems/block)
  for (int it = 0; it < 16; ++it) {
    const int q = q0 + it * 256 + tid;              // float4 index
    __builtin_prefetch((const float*)(cb4 + q) + 8192, 0, 1);   // -> global_prefetch_b8
    const float4 v4 = cb4[q];
    const int e = q * 4;
    const int p0 = e % HWPIX;                       // constant decomposition for all 4 lanes
    const float vv[4] = {v4.x, v4.y, v4.z, v4.w};
    #pragma unroll
    for (int k = 0; k < 4; ++k) {
      const float v = vv[k];
      if (v > 0.05f) {
        float s = v * tb[p0 + k];
        int bin = (int)(s * 1024.0f);
        bin = bin < 0 ? 0 : (bin > BINS - 1 ? BINS - 1 : bin);
        atomicAdd(&lh[bin], 1);
      }
    }
  }
  __syncthreads();
  for (int b = tid; b < BINS; b += 256)
    if (lh[b]) atomicAdd(&ghist[img * BINS + b], lh[b]);
}

// ---------------- pass 2: find cutoff bin ----------------
__global__ void k_cutoff(const int* __restrict__ ghist, int* __restrict__ cutoff) {
  const int img = blockIdx.x;
  if (threadIdx.x == 0) {
    int run = 0, B = 0;
    for (int b = BINS - 1; b >= 0; --b) {
      run += ghist[img * BINS + b];
      if (run >= KREAL) { B = b; break; }
    }
    cutoff[img] = B;
  }
}

// ---------------- pass 3: gather candidates >= cutoff bin (float4 streaming) ----------------
__global__ void k_gather(const float* __restrict__ cls,
                         const float* __restrict__ ctr,
                         const int* __restrict__ cutoff,
                         int* __restrict__ counter,
                         unsigned long long* __restrict__ cand) {
  const int img = blockIdx.y;
  const int tid = threadIdx.x;
  const int B = cutoff[img];
  const float4* cb4 = (const float4*)(cls + (size_t)img * CHW);
  const float*  tb  = ctr + (size_t)img * HWPIX;
  const int q0 = blockIdx.x * 4096;
  for (int it = 0; it < 16; ++it) {
    const int q = q0 + it * 256 + tid;
    const float4 v4 = cb4[q];
    const int e = q * 4;
    const int p0 = e % HWPIX;
    const int c  = e / HWPIX;                       // same class for all 4 (HWPIX % 4 == 0)
    const float vv[4] = {v4.x, v4.y, v4.z, v4.w};
    #pragma unroll
    for (int k = 0; k < 4; ++k) {
      const float v = vv[k];
      if (v > 0.05f) {
        float s = v * tb[p0 + k];
        int bin = (int)(s * 1024.0f);
        bin = bin < 0 ? 0 : (bin > BINS - 1 ? BINS - 1 : bin);
        if (bin >= B) {
          const int pos = atomicAdd(&counter[img], 1);
          if (pos < CAP) {
            const unsigned f = (unsigned)((p0 + k) * NCLS + c);  // reference flat index
            cand[(size_t)img * CAP + pos] =
                ((unsigned long long)__float_as_uint(s) << 32) |
                (unsigned long long)(0xFFFFFFFFu - f);           // desc score, asc idx
          }
        }
      }
    }
  }
}

// ---------------- pass 4: exact top-1000 via in-LDS bitonic sort ----------------
__global__ void k_sort(const unsigned long long* __restrict__ cand,
                       const int* __restrict__ counter,
                       float* __restrict__ tscore, int* __restrict__ tidx) {
  __shared__ unsigned long long keys[CAP];                // 32 KB LDS
  const int img = blockIdx.x, tid = threadIdx.x;          // 1024 threads
  int count = counter[img]; if (count > CAP) count = CAP;
  for (int e = tid; e < CAP; e += 1024)
    keys[e] = (e < count) ? cand[(size_t)img * CAP + e] : 0ULL;
  __syncthreads();
  for (int k = 2; k <= CAP; k <<= 1) {
    for (int j = k >> 1; j > 0; j >>= 1) {
      for (int i = tid; i < CAP; i += 1024) {
        const int ix = i ^ j;
        if (ix > i) {
          const bool desc = ((i & k) == 0);
          unsigned long long A = keys[i], Bv = keys[ix];
          const bool sw = desc ? (A < Bv) : (A > Bv);
          if (sw) { keys[i] = Bv; keys[ix] = A; }
        }
      }
      __syncthreads();
    }
  }
  const int r = tid;
  float sc = NEGV; int id = 0;
  if (r < KREAL && r < count) {
    const unsigned long long kk = keys[r];
    sc = __uint_as_float((unsigned)(kk >> 32));
    id = (int)(0xFFFFFFFFu - (unsigned)(kk & 0xFFFFFFFFull));
  }
  tscore[img * KPAD + r] = sc;
  tidx[img * KPAD + r] = id;
}

// ---------------- pass 5: decode + clip boxes ----------------
__global__ void k_decode(const float* __restrict__ tscore, const int* __restrict__ tidx,
                         const float* __restrict__ locations,
                         const float* __restrict__ reg,
                         const int* __restrict__ image_sizes,
                         float* __restrict__ bx1, float* __restrict__ by1,
                         float* __restrict__ bx2, float* __restrict__ by2,
                         float* __restrict__ area, int* __restrict__ label,
                         float* __restrict__ dscore) {
  const int img = blockIdx.x, r = threadIdx.x;            // 1024 threads
  const int g = img * KPAD + r;
  const float s = tscore[g];
  float X1 = 0.f, Y1 = 0.f, X2 = 0.f, Y2 = 0.f, ar = 0.f, ds = 0.f;
  int lab = 0;
  if (s > NEGV * 0.5f) {
    const int f = tidx[g];
    const int p = f / NCLS, c = f % NCLS;
    lab = c + 1;
    const float lx = locations[p * 2 + 0];
    const float ly = locations[p * 2 + 1];
    const size_t rb = (size_t)img * 4 * HWPIX + p;
    const float l = reg[rb], t = reg[rb + HWPIX];
    const float rr = reg[rb + 2 * (size_t)HWPIX], bb = reg[rb + 3 * (size_t)HWPIX];
    const float hi = (float)image_sizes[img * 2 + 0] - 1.0f;
    const float wi = (float)image_sizes[img * 2 + 1] - 1.0f;
    X1 = fminf(fmaxf(lx - l, 0.f), wi);
    X2 = fminf(fmaxf(lx + rr, 0.f), wi);
    Y1 = fminf(fmaxf(ly - t, 0.f), hi);
    Y2 = fminf(fmaxf(ly + bb, 0.f), hi);
    ar = fmaxf(X2 - X1, 0.f) * fmaxf(Y2 - Y1, 0.f);
    ds = sqrtf(s);
  }
  bx1[g] = X1; by1[g] = Y1; bx2[g] = X2; by2[g] = Y2;
  area[g] = ar; label[g] = lab; dscore[g] = ds;
}

// ---------------- pass 6: pairwise IoU adjacency (WMMA outer-sum) ----------------
// One wave per 16x16 tile. union = (area_i + area_j) - inter, where the
// outer-sum tile is computed by V_WMMA_F32_16X16X4_F32 as a rank-2 matmul.
// The IoU test is done division-free: inter > TH * max(union, 1e-9).
__global__ void k_pairs(const float* __restrict__ bx1, const float* __restrict__ by1,
                        const float* __restrict__ bx2, const float* __restrict__ by2,
                        const float* __restrict__ area, const int* __restrict__ label,
                        unsigned int* __restrict__ adj) {
  __shared__ float sI[8][16][4];
  __shared__ float sJ[8][16][4];
  const int img  = blockIdx.y;
  const int w    = threadIdx.x >> 5;
  const int lane = threadIdx.x & 31;
  const int tile = blockIdx.x * 8 + w;                    // 512 blocks x * 8 waves = 4096 tiles
  const int ti = tile >> 6, tj = tile & 63;
  const int base = img * KPAD;

  // stage class-offset box coords (lanes 0..15 -> rows, 16..31 -> cols)
  {
    const int sel = (lane < 16) ? (ti * 16 + lane) : (tj * 16 + (lane - 16));
    const int g = base + sel;
    const float off = (float)label[g] * CLSOFF;
    const float x1 = bx1[g] + off, y1 = by1[g] + off;
    const float x2 = bx2[g] + off, y2 = by2[g] + off;
    if (lane < 16) {
      sI[w][lane][0] = x1; sI[w][lane][1] = y1; sI[w][lane][2] = x2; sI[w][lane][3] = y2;
    } else {
      sJ[w][lane - 16][0] = x1; sJ[w][lane - 16][1] = y1;
      sJ[w][lane - 16][2] = x2; sJ[w][lane - 16][3] = y2;
    }
  }
  // WMMA operands: A(16x4)=[area_i,1,0,0], B(4x16)=[1;area_j;0;0]
  v2f a, b;
  a.x = (lane < 16) ? area[base + ti * 16 + lane] : 0.0f;  // A: K=0 (lanes<16), K=2 (else)
  a.y = (lane < 16) ? 1.0f : 0.0f;                         // A: K=1 / K=3
  b.x = (lane < 16) ? 1.0f : 0.0f;                         // B: K=0 row / K=2 row
  b.y = (lane < 16) ? area[base + tj * 16 + lane] : 0.0f;  // B: K=1 row / K=3 row
  v8f c = {0.f, 0.f, 0.f, 0.f, 0.f, 0.f, 0.f, 0.f};
  __syncthreads();                                         // EXEC all-ones from here

  v8f usum;
#if __has_builtin(__builtin_amdgcn_wmma_f32_16x16x4_f32)
  usum = __builtin_amdgcn_wmma_f32_16x16x4_f32(
      /*neg_a=*/false, a, /*neg_b=*/false, b,
      /*c_mod=*/(short)0, c, /*reuse_a=*/false, /*reuse_b=*/false);
#else
  #pragma unroll
  for (int r = 0; r < 8; ++r)
    usum[r] = area[base + ti * 16 + r + ((lane >> 4) << 3)] +
              area[base + tj * 16 + (lane & 15)];
#endif

  const int nl = lane & 15;
  const float jx1 = sJ[w][nl][0], jy1 = sJ[w][nl][1];
  const float jx2 = sJ[w][nl][2], jy2 = sJ[w][nl][3];
  const int n = tj * 16 + nl;
  #pragma unroll
  for (int r = 0; r < 8; ++r) {
    const int ml = r + ((lane >> 4) << 3);                 // D layout: M = r + 8*(lane>=16)
    const int m = ti * 16 + ml;
    const float ix1 = sI[w][ml][0], iy1 = sI[w][ml][1];
    const float ix2 = sI[w][ml][2], iy2 = sI[w][ml][3];
    const float xx1 = fmaxf(ix1, jx1), yy1 = fmaxf(iy1, jy1);
    const float xx2 = fminf(ix2, jx2), yy2 = fminf(iy2, jy2);
    const float inter = fmaxf(xx2 - xx1, 0.f) * fmaxf(yy2 - yy1, 0.f);
    const float uni = fmaxf(usum[r] - inter, 1e-9f);
    // iou > TH  <=>  inter > TH * uni   (uni > 0), division-free
    if (n > m && inter > NMSTH * uni)
      atomicOr(&adj[((size_t)base + m) * 32 + (n >> 5)], 1u << (n & 31));
  }
}

// ---------------- pass 7: greedy NMS scan + final top-100 emit ----------------
__global__ void k_nms(const unsigned int* __restrict__ adj, const int* __restrict__ label,
                      const float* __restrict__ bx1, const float* __restrict__ by1,
                      const float* __restrict__ bx2, const float* __restrict__ by2,
                      const float* __restrict__ dscore, float* __restrict__ out) {
  __shared__ unsigned int supp[32];
  __shared__ unsigned char keep[KPAD];
  __shared__ int fi[OUTN];
  __shared__ unsigned char fk[OUTN];
  __shared__ int sKeep;
  const int img = blockIdx.x, tid = threadIdx.x;          // 128 threads
  const int base = img * KPAD;
  if (tid < 32) {
    unsigned int msk = 0;
    for (int b = 0; b < 32; ++b)
      if (label[base + tid * 32 + b] == 0) msk |= (1u << b);   // invalid -> pre-suppressed
    supp[tid] = msk;
  }
  __syncthreads();
  for (int i = 0; i < KPAD; ++i) {
    if (tid == 0) {
      const int k = ((supp[i >> 5] >> (i & 31)) & 1u) ? 0 : 1;
      keep[i] = (unsigned char)k;
      sKeep = k;
    }
    __syncthreads();
    if (sKeep && tid < 32) supp[tid] |= adj[((size_t)base + i) * 32 + tid];
    __syncthreads();
  }
  if (tid == 0) {
    int cnt = 0;
    for (int i = 0; i < KPAD && cnt < OUTN; ++i)
      if (keep[i]) { fi[cnt] = i; fk[cnt] = 1; ++cnt; }
    for (int i = 0; i < KPAD && cnt < OUTN; ++i)
      if (!keep[i]) { fi[cnt] = i; fk[cnt] = 0; ++cnt; }
  }
  __syncthreads();
  if (tid < OUTN) {
    const int i = fi[tid], g = base + i, k = fk[tid];
    float* ob = out;                          // boxes  [0, 6400)
    float* os = out + NIMG * OUTN * 4;        // scores [6400, 8000)
    float* ol = os + NIMG * OUTN;             // labels [8000, 9600)
    float* ov = ol + NIMG * OUTN;             // valid  [9600, 11200)
    const int o = img * OUTN + tid;
    ob[o * 4 + 0] = bx1[g]; ob[o * 4 + 1] = by1[g];
    ob[o * 4 + 2] = bx2[g]; ob[o * 4 + 3] = by2[g];
    os[o] = k ? dscore[g] : 0.f;
    ol[o] = k ? (float)label[g] : 0.f;
    ov[o] = k ? 1.f : 0.f;
  }
}

// ---------------- host-side launch ----------------
extern "C" void kernel_launch(void* const* d_in, const int* in_sizes, int n_in,
                              void* d_out, int out_size, void* d_ws, size_t ws_size,
                              hipStream_t stream) {
  (void)in_sizes; (void)n_in; (void)out_size; (void)ws_size;
  const float* locations   = (const float*)d_in[0];
  const float* box_cls     = (const float*)d_in[1];
  const float* box_reg     = (const float*)d_in[2];
  const float* centerness  = (const float*)d_in[3];
  const int*   image_sizes = (const int*)d_in[4];
  float* out = (float*)d_out;

  char* ws = (char*)d_ws;
  size_t off = 0;
  auto wsa = [&](size_t bytes) { char* p = ws + off; off += (bytes + 255) & ~(size_t)255; return p; };
  int*                ghist   = (int*)wsa((size_t)NIMG * BINS * 4);
  int*                cutoff  = (int*)wsa(NIMG * 4);
  int*                counter = (int*)wsa(NIMG * 4);
  unsigned long long* cand    = (unsigned long long*)wsa((size_t)NIMG * CAP * 8);
  float*              tscore  = (float*)wsa((size_t)NIMG * KPAD * 4);
  int*                tidx    = (int*)wsa((size_t)NIMG * KPAD * 4);
  float*              bx1     = (float*)wsa((size_t)NIMG * KPAD * 4);
  float*              by1     = (float*)wsa((size_t)NIMG * KPAD * 4);
  float*              bx2     = (float*)wsa((size_t)NIMG * KPAD * 4);
  float*              by2     = (float*)wsa((size_t)NIMG * KPAD * 4);
  float*              area    = (float*)wsa((size_t)NIMG * KPAD * 4);
  float*              dscore  = (float*)wsa((size_t)NIMG * KPAD * 4);
  int*                label   = (int*)wsa((size_t)NIMG * KPAD * 4);
  unsigned int*       adj     = (unsigned int*)wsa((size_t)NIMG * KPAD * 32 * 4);

  // per-call zeroing (graph-capture safe)
  hipMemsetAsync(ghist,   0, (size_t)NIMG * BINS * 4, stream);
  hipMemsetAsync(counter, 0, NIMG * 4, stream);
  hipMemsetAsync(adj,     0, (size_t)NIMG * KPAD * 32 * 4, stream);

  k_hist  <<<dim3(250, NIMG), 256, 0, stream>>>(box_cls, centerness, ghist);
  k_cutoff<<<NIMG, 32, 0, stream>>>(ghist, cutoff);
  k_gather<<<dim3(250, NIMG), 256, 0, stream>>>(box_cls, centerness, cutoff, counter, cand);
  k_sort  <<<NIMG, 1024, 0, stream>>>(cand, counter, tscore, tidx);
  k_decode<<<NIMG, 1024, 0, stream>>>(tscore, tidx, locations, box_reg, image_sizes,
                                      bx1, by1, bx2, by2, area, label, dscore);
  k_pairs <<<dim3(512, NIMG), 256, 0, stream>>>(bx1, by1, bx2, by2, area, label, adj);
  k_nms   <<<NIMG, 128, 0, stream>>>(adj, label, bx1, by1, bx2, by2, dscore, out);
}